// BilateralSliceApply_51092930953398
// MI455X (gfx1250) — compile-verified
//
#include <hip/hip_runtime.h>
#include <stdint.h>

// Problem sizes (fixed by setup_inputs in the reference)
#define BB   4
#define GH   16
#define GW   16
#define GD   8
#define NC   12            // n_out * (n_in + 1) = 3 * 4
#define HH   1024
#define WW   1024
#define NIN  3

#define ROW_FLOATS (GW * GD * NC)   // 1536 floats = 6144 bytes per (b, gy) grid row
#define ROW_CHUNKS (ROW_FLOATS / 4) // 384 16-byte chunks per grid row
#define TOT_CHUNKS (2 * ROW_CHUNKS) // 768 chunks staged (two y-rows), 12 KB LDS

__device__ __forceinline__ void fma4(float4& a, float w, const float4& c) {
    a.x = fmaf(w, c.x, a.x);
    a.y = fmaf(w, c.y, a.y);
    a.z = fmaf(w, c.z, a.z);
    a.w = fmaf(w, c.w, a.w);
}

__global__ __launch_bounds__(256)
void bslice_apply_kernel(const float* __restrict__ grid,
                         const float* __restrict__ guide,
                         const float* __restrict__ inp,
                         float* __restrict__ out) {
    // 12KB staged grid slab: [yrow(2)][gx(16)][gz(8)][c(12)]
    __shared__ float lds[2 * GW * GD * NC];

    const int t  = threadIdx.x;
    const int by = blockIdx.x;        // b*HH + y
    const int b  = by >> 10;
    const int y  = by & (HH - 1);

    // ---- y taps (uniform per block) ----
    const float gy  = (y + 0.5f) * (1.0f / 64.0f);   // * (GH/HH)
    const float fy  = floorf(gy - 0.5f);
    const float wy0 = fmaxf(1.0f - fabsf(fy + 0.5f - gy), 0.0f);
    const float wy1 = fmaxf(1.0f - fabsf(fy + 1.5f - gy), 0.0f);
    const int   yi0 = min(max((int)fy,     0), GH - 1);
    const int   yi1 = min(max((int)fy + 1, 0), GH - 1);

    // ---- async stage of grid rows yi0, yi1 into LDS (CDNA5 async-to-LDS path) ----
    {
        const uint64_t gbase = (uint64_t)(uintptr_t)grid;
        const uint64_t rb0 = gbase + (uint64_t)((b * GH + yi0) * ROW_FLOATS) * 4ull;
        const uint64_t rb1 = gbase + (uint64_t)((b * GH + yi1) * ROW_FLOATS) * 4ull;
        const uint32_t lds_base = (uint32_t)(uintptr_t)(&lds[0]); // low 32 bits = LDS offset
#pragma unroll
        for (int k = 0; k < 3; ++k) {
            const int j    = t + 256 * k;          // chunk id 0..767 (16B chunks)
            const int row  = (j >= ROW_CHUNKS) ? 1 : 0;
            const int joff = j - row * ROW_CHUNKS;
            const uint64_t ga = (row ? rb1 : rb0) + (uint64_t)joff * 16ull;
            const uint32_t la = lds_base + (uint32_t)j * 16u;
            asm volatile("global_load_async_to_lds_b128 %0, %1, off"
                         :: "v"(la), "v"(ga) : "memory");
        }
        asm volatile("s_wait_asynccnt 0x0" ::: "memory");
        __syncthreads();
    }

    const float4* __restrict__ lds4 = (const float4*)lds;

    // ---- 4 pixels per thread, coalesced along x ----
#pragma unroll
    for (int k = 0; k < 4; ++k) {
        const int x   = t + 256 * k;
        const int pix = by * WW + x;

        const float gv = guide[pix];
        const float* ip = inp + pix * NIN;
        const float r  = ip[0];
        const float g_ = ip[1];
        const float bl = ip[2];

        // x taps
        const float gx  = (x + 0.5f) * (1.0f / 64.0f);   // * (GW/WW)
        const float fx  = floorf(gx - 0.5f);
        const float wx0 = fmaxf(1.0f - fabsf(fx + 0.5f - gx), 0.0f);
        const float wx1 = fmaxf(1.0f - fabsf(fx + 1.5f - gx), 0.0f);
        const int   xi0 = min(max((int)fx,     0), GW - 1);
        const int   xi1 = min(max((int)fx + 1, 0), GW - 1);

        // z taps (guide-dependent)
        const float gz  = gv * (float)GD;
        const float fz  = floorf(gz - 0.5f);
        const float wz0 = fmaxf(1.0f - fabsf(fz + 0.5f - gz), 0.0f);
        const float wz1 = fmaxf(1.0f - fabsf(fz + 1.5f - gz), 0.0f);
        const int   zi0 = min(max((int)fz,     0), GD - 1);
        const int   zi1 = min(max((int)fz + 1, 0), GD - 1);

        const float wys[2] = {wy0, wy1};
        const int   xis[2] = {xi0, xi1};
        const float wxs[2] = {wx0, wx1};
        const int   zis[2] = {zi0, zi1};
        const float wzs[2] = {wz0, wz1};

        float4 acc0 = {0.f, 0.f, 0.f, 0.f};  // coeff c0..c3   (out0: *r,*g,*b,+bias)
        float4 acc1 = {0.f, 0.f, 0.f, 0.f};  // coeff c4..c7   (out1)
        float4 acc2 = {0.f, 0.f, 0.f, 0.f};  // coeff c8..c11  (out2)

#pragma unroll
        for (int ty = 0; ty < 2; ++ty) {
#pragma unroll
            for (int tx = 0; tx < 2; ++tx) {
                const float wyx = wys[ty] * wxs[tx];
                const int byx = (ty * GW + xis[tx]) * GD;
#pragma unroll
                for (int tz = 0; tz < 2; ++tz) {
                    const float w = wyx * wzs[tz];
                    const int base = (byx + zis[tz]) * 3;   // float4 units (12 floats/cell)
                    fma4(acc0, w, lds4[base + 0]);
                    fma4(acc1, w, lds4[base + 1]);
                    fma4(acc2, w, lds4[base + 2]);
                }
            }
        }

        float* op = out + pix * NIN;
        op[0] = fmaf(acc0.x, r, fmaf(acc0.y, g_, fmaf(acc0.z, bl, acc0.w)));
        op[1] = fmaf(acc1.x, r, fmaf(acc1.y, g_, fmaf(acc1.z, bl, acc1.w)));
        op[2] = fmaf(acc2.x, r, fmaf(acc2.y, g_, fmaf(acc2.z, bl, acc2.w)));
    }
}

extern "C" void kernel_launch(void* const* d_in, const int* in_sizes, int n_in,
                              void* d_out, int out_size, void* d_ws, size_t ws_size,
                              hipStream_t stream) {
    const float* grid  = (const float*)d_in[0];  // (4,16,16,8,12) f32
    const float* guide = (const float*)d_in[1];  // (4,1024,1024)  f32
    const float* inp   = (const float*)d_in[2];  // (4,1024,1024,3) f32
    float* out = (float*)d_out;                  // (4,1024,1024,3) f32

    (void)in_sizes; (void)n_in; (void)out_size; (void)d_ws; (void)ws_size;

    dim3 grid_dim(BB * HH);   // 4096 blocks: one per (b, y) row
    dim3 block_dim(256);      // 8 waves (wave32); 4 pixels per thread
    bslice_apply_kernel<<<grid_dim, block_dim, 0, stream>>>(grid, guide, inp, out);
}